// KGReasoning_3384434230128
// MI455X (gfx1250) — compile-verified
//
#include <hip/hip_runtime.h>
#include <hip/hip_bf16.h>

#define NENTITY 50000
#define NRELATION 500
#define DIM 128
#define NBASE 30
#define BQ 16
#define TWO_DIM 256
#define GAMMA_F 12.0f
#define CEN_F 0.25f
#define LN_EPS_F 1e-5f
#define PI_F 3.14159265358979323846f
#define EMB_RANGE_F 0.109375f
#define ANG (PI_F / EMB_RANGE_F)

typedef float v2f __attribute__((ext_vector_type(2)));
typedef float v8f __attribute__((ext_vector_type(8)));

// ---------------------------------------------------------------------------
// Kernel 0: zero the split-K accumulator (harness does not re-poison ws)
// ---------------------------------------------------------------------------
__global__ void kgr_zero(float* __restrict__ p, int n) {
    int i = blockIdx.x * blockDim.x + threadIdx.x;
    if (i < n) p[i] = 0.0f;
}

// ---------------------------------------------------------------------------
// Kernel 1: split-K projection GEMM with V_WMMA_F32_16X16X4_F32.
//   C[m,o] += sum_{i<128} (att[m,r]*e[m,i]) * rel_base[r,i,o]   (one r per WG)
// grid = 30 bases * 4 N-chunks; block = 256 (8 waves, one 16x16 tile each).
// ---------------------------------------------------------------------------
__global__ __launch_bounds__(256) void kgr_gemm_wmma(
    const float* __restrict__ emb, const float* __restrict__ rel_att,
    const float* __restrict__ rel_base,
    const int* __restrict__ h_idx, const int* __restrict__ r_idx,
    float* __restrict__ out_pre) {
    __shared__ float e_lds[32][DIM];   // 16 KB: e[m,i] = pi*tanh(emb*ANG)
    __shared__ float att_lds[32];

    const int tid     = threadIdx.x;
    const int r       = blockIdx.x >> 2;   // base index 0..29
    const int n_chunk = blockIdx.x & 3;    // 64-col chunk of N=256

    for (int idx = tid; idx < 32 * DIM; idx += 256) {
        int m = idx >> 7, i = idx & 127;
        float v = emb[(long)h_idx[m] * DIM + i];
        e_lds[m][i] = PI_F * tanhf(v * ANG);
    }
    if (tid < 32)
        att_lds[tid] = PI_F * tanhf(rel_att[(long)r_idx[tid] * NBASE + r] * ANG);
    __syncthreads();

    const int lane   = tid & 31;
    const int wave   = tid >> 5;
    const int m_base = (wave >> 2) << 4;                       // 0 or 16
    const int n_col  = n_chunk * 64 + (wave & 3) * 16 + (lane & 15);
    const int m_row  = m_base + (lane & 15);
    const int khalf  = (lane >> 4) << 1;                       // 0 or 2
    const float attv = att_lds[m_row];
    const float* Bp  = rel_base + (long)r * TWO_DIM * TWO_DIM + n_col;

    v8f c = {0.f, 0.f, 0.f, 0.f, 0.f, 0.f, 0.f, 0.f};
#pragma unroll 4
    for (int k = 0; k < DIM; k += 4) {
        const int kk = k + khalf;
        v2f a, b;
        // A 16x4 layout: lanes 0-15 hold K={k,k+1}, lanes 16-31 hold K={k+2,k+3}
        a.x = attv * e_lds[m_row][kk];
        a.y = attv * e_lds[m_row][kk + 1];
        // B 4x16 layout: rows striped across lanes, half-wave offset of 2 rows
        b.x = Bp[(long)kk * TWO_DIM];
        b.y = Bp[(long)(kk + 1) * TWO_DIM];
        c = __builtin_amdgcn_wmma_f32_16x16x4_f32(false, a, false, b,
                                                  (short)0, c, false, false);
    }

    // C/D layout: VGPR v -> M = v (lanes 0-15) or v+8 (lanes 16-31), N = lane&15
    const int m_hi = (lane >> 4) << 3;
#pragma unroll
    for (int v = 0; v < 8; ++v) {
        int row = m_base + v + m_hi;
        atomicAdd(&out_pre[row * TWO_DIM + n_col], c[v]);
    }
}

// ---------------------------------------------------------------------------
// Kernel 2: bias + layernorm + tanh conversions + cone intersection + pack
// Packs per-(q,d): {axis, sin(arg/2), cos(arg/2), |sin(arg/2)|} as float4.
// ---------------------------------------------------------------------------
__global__ __launch_bounds__(256) void kgr_finalize(
    const float* __restrict__ rel_att, const float* __restrict__ rel_bias,
    const int* __restrict__ r_idx,
    const float* __restrict__ out_pre, float4* __restrict__ qpack) {
    __shared__ float att_lds[32][NBASE];
    __shared__ float ax_lds[32][DIM];
    __shared__ float ag_lds[32][DIM];
    __shared__ float2 red[256];
    const int tid = threadIdx.x;

    if (tid < 32) {
        for (int rr = 0; rr < NBASE; ++rr)
            att_lds[tid][rr] =
                PI_F * tanhf(rel_att[(long)r_idx[tid] * NBASE + rr] * ANG);
    }
    __syncthreads();

    for (int m = 0; m < 32; ++m) {
        float val = out_pre[m * TWO_DIM + tid];
#pragma unroll 5
        for (int rr = 0; rr < NBASE; ++rr)
            val += att_lds[m][rr] * rel_bias[rr * TWO_DIM + tid];

        red[tid] = make_float2(val, val * val);
        __syncthreads();
        for (int s = 128; s > 0; s >>= 1) {
            if (tid < s) {
                red[tid].x += red[tid + s].x;
                red[tid].y += red[tid + s].y;
            }
            __syncthreads();
        }
        const float mu  = red[0].x * (1.0f / 256.0f);
        const float var = red[0].y * (1.0f / 256.0f) - mu * mu;
        const float nv  = (val - mu) * rsqrtf(var + LN_EPS_F);
        if (tid < DIM)
            ax_lds[m][tid] = PI_F * tanhf(nv * ANG);
        else
            ag_lds[m][tid - DIM] = 0.5f * PI_F * tanhf(2.0f * nv * ANG) + 0.5f * PI_F;
        __syncthreads();
    }

    for (int p = tid; p < BQ * DIM; p += 256) {
        const int j = p >> 7, d = p & 127;
        const float axis1 = ax_lds[j][d],      arg1 = ag_lds[j][d];
        const float axis2 = ax_lds[16 + j][d], arg2 = ag_lds[16 + j][d];
        const float up1 = axis1 + arg1, lo1 = axis1 - arg1;
        const float up2 = axis2 + arg2, lo2 = axis2 - arg2;
        const bool m11 = (up1 >= up2) && (up2 >= lo1) && (lo1 >= lo2);
        const bool m12 = (up1 >= up2) && (up2 >= lo2) && (lo2 > lo1);
        const bool m13 = (up1 >= lo1) && (lo1 > up2) && (up2 >= lo2);
        const bool m21 = (up2 >= up1) && (up1 >= lo2) && (lo2 >= lo1);
        const bool m22 = (up2 >= up1) && (up1 >= lo1) && (lo1 > lo2);
        const bool m23 = (up2 >= lo2) && (lo2 > up1) && (up1 >= lo1);
        float arg_i = fminf(arg1, arg2);
        if (m11) arg_i = fabsf(up2 - lo1) * 0.5f;
        if (m12) arg_i = arg2;
        if (m13) arg_i = 0.0f;
        if (m21) arg_i = fabsf(up1 - lo2) * 0.5f;
        if (m22) arg_i = arg1;
        if (m23) arg_i = 0.0f;
        float axis_i = fminf(axis1, axis2);
        if (m11) axis_i = up2 - arg_i;
        if (m12) axis_i = axis2;
        if (m13) axis_i = 0.5f * lo1 + 0.5f * up2;   // delta_none = 0.5
        if (m21) axis_i = up1 - arg_i;
        if (m22) axis_i = axis1;
        if (m23) axis_i = 0.5f * lo2 + 0.5f * up1;

        const float hv = 0.5f * arg_i;
        const float sv = __sinf(hv), cv = __cosf(hv);
        qpack[p] = make_float4(axis_i, sv, cv, fabsf(sv));
    }
}

// ---------------------------------------------------------------------------
// Kernel 3: cone_logit over all entities. 2 trans + 1 ds_load_b128 per (d,q).
// ---------------------------------------------------------------------------
__global__ __launch_bounds__(256) void kgr_logits(
    const float* __restrict__ emb, const float4* __restrict__ qpack,
    float* __restrict__ out) {
    __shared__ float4 q_lds[BQ * DIM];   // 32 KB, broadcast reads
    const int tid = threadIdx.x;
    for (int p = tid; p < BQ * DIM; p += 256) q_lds[p] = qpack[p];
    __syncthreads();

    const int n = blockIdx.x * 256 + tid;
    if (n >= NENTITY) return;
    const float* ep = emb + (long)n * DIM;

    float acc[BQ];
#pragma unroll
    for (int q = 0; q < BQ; ++q) acc[q] = 0.0f;

    for (int d = 0; d < DIM; ++d) {
        const float th = PI_F * tanhf(ep[d] * ANG);
#pragma unroll
        for (int q = 0; q < BQ; ++q) {
            const float4 pk = q_lds[q * DIM + d];   // {a, sin(g/2), cos(g/2), base}
            const float u  = (th - pk.x) * 0.5f;
            const float su = __sinf(u);
            const float cu = __cosf(u);
            const float s0 = fabsf(su);
            const float t1 = su * pk.z;
            const float t2 = cu * pk.y;
            const float s1 = fabsf(t1 + t2);        // |sin((th-(a-g))/2)|
            const float s2 = fabsf(t1 - t2);        // |sin((th-(a+g))/2)|
            float dout = fminf(s1, s2);
            dout = (s0 < pk.w) ? 0.0f : dout;
            const float din = fminf(s0, pk.w);
            acc[q] += dout + CEN_F * din;
        }
    }
#pragma unroll
    for (int q = 0; q < BQ; ++q)
        out[(long)q * NENTITY + n] = GAMMA_F - acc[q];
}

// ---------------------------------------------------------------------------
extern "C" void kernel_launch(void* const* d_in, const int* in_sizes, int n_in,
                              void* d_out, int out_size, void* d_ws, size_t ws_size,
                              hipStream_t stream) {
    const float* emb      = (const float*)d_in[0];   // (50000,128)
    const float* rel_att  = (const float*)d_in[1];   // (500,30)
    const float* rel_base = (const float*)d_in[2];   // (30,256,256)
    const float* rel_bias = (const float*)d_in[3];   // (30,256)
    const int*   h_idx    = (const int*)d_in[4];     // (2,16)
    const int*   r_idx    = (const int*)d_in[5];     // (2,16)
    float* out = (float*)d_out;                      // (16,50000)

    float*  out_pre = (float*)d_ws;                              // 32*256 f32
    float4* qpack   = (float4*)((char*)d_ws + 32 * TWO_DIM * sizeof(float));

    kgr_zero<<<(32 * TWO_DIM + 255) / 256, 256, 0, stream>>>(out_pre, 32 * TWO_DIM);
    kgr_gemm_wmma<<<NBASE * 4, 256, 0, stream>>>(emb, rel_att, rel_base,
                                                 h_idx, r_idx, out_pre);
    kgr_finalize<<<1, 256, 0, stream>>>(rel_att, rel_bias, r_idx, out_pre, qpack);
    kgr_logits<<<(NENTITY + 255) / 256, 256, 0, stream>>>(emb, qpack, out);
}